// BackboneHBonds_40819369181342
// MI455X (gfx1250) — compile-verified
//
#include <hip/hip_runtime.h>
#include <hip/hip_bf16.h>

// BackboneHBonds — gather/streaming memory-bound kernel for MI455X (gfx1250).
// B=4, N=16384, K=64 fixed by the reference setup.
// HBM roofline ~3us; X (3MB) and C (0.25MB) live in the 192MB L2 so the
// per-edge gathers are L2 hits. No matrix structure -> no WMMA by design.

namespace {
constexpr int   Bc      = 4;
constexpr int   Nc      = 16384;   // power of two: row>>14 = b, row&(Nc-1) = i
constexpr int   Kc      = 64;      // power of two: t>>6 = row, t&63 = k
constexpr float EPSc    = 0.001f;
constexpr float COEFFc  = 0.42f * 0.2f * 332.0f;   // 27.888
constexpr float LEN_NHc = 1.015f;
constexpr float CUT_Ec  = -0.5f;
constexpr float CUT_Dc  = 3.6f;

__device__ __forceinline__ float3 sub3(float3 a, float3 b) {
  return make_float3(a.x - b.x, a.y - b.y, a.z - b.z);
}
__device__ __forceinline__ float3 add3(float3 a, float3 b) {
  return make_float3(a.x + b.x, a.y + b.y, a.z + b.z);
}
__device__ __forceinline__ float3 nrm(float3 v) {
  // reference: v / sqrt(dot(v,v) + eps)  -> v * rsqrt(dot+eps)
  float inv = rsqrtf(fmaf(v.x, v.x, fmaf(v.y, v.y, v.z * v.z)) + EPSc);
  return make_float3(v.x * inv, v.y * inv, v.z * inv);
}
__device__ __forceinline__ float dist2(float3 a, float3 b) {
  float dx = a.x - b.x, dy = a.y - b.y, dz = a.z - b.z;
  return fmaf(dx, dx, fmaf(dy, dy, dz * dz));
}
} // namespace

__global__ __launch_bounds__(256) void backbone_hbonds_kernel(
    const float* __restrict__ X,   // (B,N,4,3)
    const int*   __restrict__ Cg,  // (B,N)   chain ids
    const int*   __restrict__ E,   // (B,N,K) neighbor indices
    const float* __restrict__ M,   // (B,N,K) mask_ij
    float*       __restrict__ out) // [hbonds | mask_hb | H] flat
{
  const int t = blockIdx.x * 256 + threadIdx.x;  // edge id == output index
  const int k = t & (Kc - 1);
  // row (= b*N + i) is identical across each wave32 (K=64, wave-aligned
  // blocks) -> promote to SGPR so the per-row data uses scalar loads.
  const int row = __builtin_amdgcn_readfirstlane(t >> 6);
  const int i   = row & (Nc - 1);
  const int b   = row >> 14;

  // ---------- per-row (wave-uniform, scalar-load path) ----------
  const float* xr = X + (size_t)row * 12;
  const float4 r0 = *reinterpret_cast<const float4*>(xr);      // Nx Ny Nz Ca.x
  const float2 r1 = *reinterpret_cast<const float2*>(xr + 4);  // Ca.y Ca.z
  const float* xp = X + (size_t)(row - (i > 0 ? 1 : 0)) * 12;  // prev residue (clamped at i=0)
  const float2 p0 = *reinterpret_cast<const float2*>(xp + 6);  // Cprev.x Cprev.y
  const float  p2 = xp[8];                                     // Cprev.z

  const float3 Ni = make_float3(r0.x, r0.y, r0.z);
  const float3 Ca = make_float3(r0.w, r1.x, r1.y);
  const float3 Cp = make_float3(p0.x, p0.y, p2);

  const float3 u = nrm(add3(nrm(sub3(Ni, Cp)), nrm(sub3(Ni, Ca))));
  const float3 H = make_float3(fmaf(LEN_NHc, u.x, Ni.x),
                               fmaf(LEN_NHc, u.y, Ni.y),
                               fmaf(LEN_NHc, u.z, Ni.z));
  const int Ci = Cg[row];

  // ---------- per-edge ----------
  const int   e = E[t];
  const float m = M[t];

  // neighbor residue: need atoms 2 (C_j) and 3 (O_j) = floats [6..11].
  // residue stride = 48B -> (+24B is 8-aligned, +32B is 16-aligned)
  const float* xj = X + ((size_t)b * Nc + (size_t)e) * 12;
  const float2 c0 = *reinterpret_cast<const float2*>(xj + 6);  // Cj.x Cj.y
  const float4 c1 = *reinterpret_cast<const float4*>(xj + 8);  // Cj.z Oj.xyz
  const float3 Cj = make_float3(c0.x, c0.y, c1.x);
  const float3 Oj = make_float3(c1.y, c1.z, c1.w);

  const float d2NO = dist2(Ni, Oj) + EPSc;
  const float d2NC = dist2(Ni, Cj) + EPSc;
  const float d2HC = dist2(H,  Cj) + EPSc;
  const float d2HO = dist2(H,  Oj) + EPSc;

  const float U = COEFFc * (rsqrtf(d2NO) - rsqrtf(d2NC) +
                            rsqrtf(d2HC) - rsqrtf(d2HO));

  const int Cn  = Cg[b * Nc + e];
  int dij = e - i; dij = dij < 0 ? -dij : dij;
  const bool local = (dij < 3) && (Ci == Cn);

  // NOTE: reference's C_prev = concat([0,C])[:,1:] == C, so mask_i = (C>0).
  const float mhb = m
      * (local ? 0.0f : 1.0f)
      * ((sqrtf(d2NO) < CUT_Dc) ? 1.0f : 0.0f)
      * ((Ci > 0) ? 1.0f : 0.0f)
      * ((Cn > 0) ? 1.0f : 0.0f);

  constexpr size_t BNK = (size_t)Bc * Nc * Kc;
  out[t]       = (U < CUT_Ec) ? mhb : 0.0f;   // hbonds
  out[BNK + t] = mhb;                          // mask_hb_ij

  if (k == 0) {                                // H_i (one lane per row)
    float* h = out + 2 * BNK + (size_t)row * 3;
    h[0] = H.x; h[1] = H.y; h[2] = H.z;
  }

  // CDNA5 speculative prefetch of the edge-index stream one grid-slice ahead
  // (emits global_prefetch_b8; dropped silently if translation fails).
  const int ahead = t + (1 << 16);
  if (ahead < (int)BNK) __builtin_prefetch(E + ahead, 0, 0);
}

extern "C" void kernel_launch(void* const* d_in, const int* in_sizes, int n_in,
                              void* d_out, int out_size, void* d_ws, size_t ws_size,
                              hipStream_t stream) {
  (void)in_sizes; (void)n_in; (void)out_size; (void)d_ws; (void)ws_size;
  const float* X  = (const float*)d_in[0];
  const int*   C  = (const int*)  d_in[1];
  const int*   E  = (const int*)  d_in[2];
  const float* M  = (const float*)d_in[3];
  float*       out = (float*)d_out;

  constexpr int total  = Bc * Nc * Kc;   // 4,194,304 edges
  constexpr int blockT = 256;            // 8 wave32s; 4 rows per block
  backbone_hbonds_kernel<<<total / blockT, blockT, 0, stream>>>(X, C, E, M, out);
}